// Cluster_loss_29875792511032
// MI455X (gfx1250) — compile-verified
//
#include <hip/hip_runtime.h>
#include <math.h>

// ---------------------------------------------------------------------------
// Cluster (discriminative) loss for MI455X / gfx1250.
//
// Strategy: the dominant cost is streaming 256MB of f32 features and doing a
// masked reduction  sums[n,c,k] = sum_p f[n,c,p]*onehot(gt[p]==k).
// That reduction is a GEMM  f(CxP) x onehot(PxK), which we feed to the exact-
// precision CDNA5 matrix op V_WMMA_F32_16X16X4_F32 (f32 in, f32 accumulate).
// Per 4-pixel group: 5 WMMAs (sums lo/hi channels, sum-of-squares lo/hi,
// counts via constant ones-row A).  f2[n,k] = column-sum over c of the sq
// tile, folded in at the end (sum over p and c commute).
//
// Phase 1: 512 WGs x 256 thr. Each WG stages 32ch x 256px tiles to LDS with
//          coalesced float4 loads, 8 waves run WMMA on 32-px slices, partial
//          (sums 512 + sq 512 + cnt 16 = 1040 f32) written to d_ws.
// Phase 2: 8 WGs reduce 64 partials/image, compute cluster means -> d_out[4..].
// Phase 3: 1 WG computes variance/distance/norm hinges + total -> d_out[0..3].
// All reductions are fixed-order (deterministic; no float atomics).
// ---------------------------------------------------------------------------

typedef float v2f __attribute__((ext_vector_type(2)));
typedef float v8f __attribute__((ext_vector_type(8)));

#define N_IMG        8
#define C_CH         32
#define P_PIX        (512*512)
#define K_CL         16
#define WGS_PER_IMG  64
#define TOTAL_WGS    (N_IMG*WGS_PER_IMG)
#define TILE_PX      256
#define TILES_PER_WG 16            // 16*256 = 4096 px per WG; 64 WGs cover 262144
#define LSTRIDE      260           // LDS dwords per channel row (bank-friendly, 16B aligned)
#define PART_STRIDE  1040          // 512 sums + 512 sq + 16 cnt

#define DELTA_VAR  0.2f
#define DELTA_DIST 0.2f
#define GAMMA_W    0.001f

// ---------------------------------------------------------------------------
__global__ __launch_bounds__(256) void cluster_phase1(
    const float* __restrict__ feat, const int* __restrict__ gt,
    float* __restrict__ partials)
{
    __shared__ float smem[10240];   // 40KB: staging (32*260=8320 f32) then 8x1280 reduction
    __shared__ int   sgt[TILE_PX];

    const int tid  = threadIdx.x;
    const int wave = tid >> 5;
    const int lane = tid & 31;
    const int half = lane >> 4;      // 0: lanes 0-15, 1: lanes 16-31
    const int col  = lane & 15;      // channel row (A) / cluster col (B,D)
    const int wg   = blockIdx.x;
    const int n    = wg / WGS_PER_IMG;
    const int jb   = wg % WGS_PER_IMG;
    const long fbase  = (long)n * C_CH * P_PIX;
    const int  pstart = jb * (TILES_PER_WG * TILE_PX);

    v8f acc0  = {};   // sums, channels 0-15   (rows=channel, cols=cluster)
    v8f acc1  = {};   // sums, channels 16-31
    v8f accq0 = {};   // sum of squares, channels 0-15
    v8f accq1 = {};   // sum of squares, channels 16-31
    v8f accc  = {};   // counts in row M=0
    const float one0 = (col == 0) ? 1.0f : 0.0f;   // A row 0 all-ones, rest zero
    v2f aone; aone.x = one0; aone.y = one0;

    for (int t = 0; t < TILES_PER_WG; ++t) {
        const int pbase = pstart + t * TILE_PX;

        // ---- stage features to LDS, fully coalesced float4 loads ----------
        #pragma unroll
        for (int pass = 0; pass < 8; ++pass) {
            const int c   = pass * 4 + (tid >> 6);
            const int px4 = (tid & 63) * 4;
            const float4 vv = *reinterpret_cast<const float4*>(
                feat + fbase + (long)c * P_PIX + pbase + px4);
            *reinterpret_cast<float4*>(&smem[c * LSTRIDE + px4]) = vv;
        }
        sgt[tid] = gt[(long)n * P_PIX + pbase + tid];
        __syncthreads();

        // prefetch next tile while we compute on this one (global_prefetch_b8)
        if (t + 1 < TILES_PER_WG) {
            #pragma unroll
            for (int pass = 0; pass < 8; ++pass) {
                const int c   = pass * 4 + (tid >> 6);
                const int px4 = (tid & 63) * 4;
                __builtin_prefetch(feat + fbase + (long)c * P_PIX + pbase + TILE_PX + px4, 0, 3);
            }
        }

        // ---- each wave: 8 groups of 4 pixels, 5 WMMAs per group -----------
        #pragma unroll
        for (int g = 0; g < 8; ++g) {
            const int p0 = wave * 32 + g * 4;
            const int pr = p0 + 2 * half;           // pixel pair this lane handles
            const int g0 = sgt[pr], g1 = sgt[pr + 1];

            // B (4x16 onehot): VGPR0 = rows {0,2}, VGPR1 = rows {1,3}
            v2f b; b.x = (g0 == col) ? 1.0f : 0.0f;
                   b.y = (g1 == col) ? 1.0f : 0.0f;

            // A (16x4): lane holds row M=col, K pair per lane-half
            v2f a0, a1;
            a0.x = smem[col * LSTRIDE + pr];
            a0.y = smem[col * LSTRIDE + pr + 1];
            a1.x = smem[(col + 16) * LSTRIDE + pr];
            a1.y = smem[(col + 16) * LSTRIDE + pr + 1];
            v2f q0, q1;
            q0.x = a0.x * a0.x; q0.y = a0.y * a0.y;
            q1.x = a1.x * a1.x; q1.y = a1.y * a1.y;

            acc0  = __builtin_amdgcn_wmma_f32_16x16x4_f32(false, a0,   false, b, (short)0, acc0,  false, false);
            acc1  = __builtin_amdgcn_wmma_f32_16x16x4_f32(false, a1,   false, b, (short)0, acc1,  false, false);
            accq0 = __builtin_amdgcn_wmma_f32_16x16x4_f32(false, q0,   false, b, (short)0, accq0, false, false);
            accq1 = __builtin_amdgcn_wmma_f32_16x16x4_f32(false, q1,   false, b, (short)0, accq1, false, false);
            accc  = __builtin_amdgcn_wmma_f32_16x16x4_f32(false, aone, false, b, (short)0, accc,  false, false);
        }
        __syncthreads();   // staging buffer reused next tile / for reduction
    }

    // ---- cross-wave reduction through LDS (fixed order, deterministic) ----
    {
        float* dst = &smem[wave * 1280 + lane];
        #pragma unroll
        for (int v = 0; v < 8; ++v) dst[0 * 256 + v * 32] = acc0[v];
        #pragma unroll
        for (int v = 0; v < 8; ++v) dst[1 * 256 + v * 32] = acc1[v];
        #pragma unroll
        for (int v = 0; v < 8; ++v) dst[2 * 256 + v * 32] = accq0[v];
        #pragma unroll
        for (int v = 0; v < 8; ++v) dst[3 * 256 + v * 32] = accq1[v];
        #pragma unroll
        for (int v = 0; v < 8; ++v) dst[4 * 256 + v * 32] = accc[v];
    }
    __syncthreads();

    for (int e = tid; e < 1280; e += 256) {
        float s = 0.0f;
        #pragma unroll
        for (int w = 0; w < 8; ++w) s += smem[w * 1280 + e];
        const int tile = e >> 8;
        const int r    = e & 255;
        const int v    = r >> 5;
        const int ln   = r & 31;
        const int hh   = ln >> 4;
        const int cc   = ln & 15;
        const int M    = v + 8 * hh;          // C/D layout: VGPR v -> rows v, v+8
        int sem; bool wr = true;
        if      (tile == 0) sem = M * 16 + cc;
        else if (tile == 1) sem = (M + 16) * 16 + cc;
        else if (tile == 2) sem = 512 + M * 16 + cc;
        else if (tile == 3) sem = 512 + (M + 16) * 16 + cc;
        else { wr = (M == 0); sem = 1024 + cc; }
        if (wr) partials[(long)wg * PART_STRIDE + sem] = s;
    }
}

// ---------------------------------------------------------------------------
__global__ __launch_bounds__(256) void cluster_phase2(
    const float* __restrict__ partials, float* __restrict__ finals,
    float* __restrict__ out)
{
    __shared__ float fin[PART_STRIDE];
    const int n   = blockIdx.x;
    const int tid = threadIdx.x;

    for (int e = tid; e < PART_STRIDE; e += 256) {
        float s = 0.0f;
        const float* p = partials + (long)n * WGS_PER_IMG * PART_STRIDE + e;
        for (int j = 0; j < WGS_PER_IMG; ++j) s += p[(long)j * PART_STRIDE];
        fin[e] = s;
        finals[n * PART_STRIDE + e] = s;
    }
    __syncthreads();

    for (int e = tid; e < 512; e += 256) {          // e = c*16 + k
        const int k = e & 15;
        const float cnt = fin[1024 + k];
        out[4 + n * 512 + e] = fin[e] / fmaxf(cnt, 1.0f);   // cluster_mean (N,C,K)
    }
}

// ---------------------------------------------------------------------------
__global__ __launch_bounds__(256) void cluster_phase3(
    const float* __restrict__ finals, float* __restrict__ out)
{
    __shared__ float redv[256], redd[256], redn[256];
    const int tid = threadIdx.x;
    float vsum = 0.0f, dsum = 0.0f, nsum = 0.0f;

    if (tid < 128) {                                 // one (n,k) per thread
        const int n = tid >> 4, k = tid & 15;
        const float* F = finals + n * PART_STRIDE;
        const float* M = out + 4 + n * 512;
        const float cnt  = F[1024 + k];
        const float safe = fmaxf(cnt, 1.0f);
        float f2 = 0.0f, cross = 0.0f, m2 = 0.0f;
        for (int c = 0; c < C_CH; ++c) {
            const float s = F[c * 16 + k];
            const float q = F[512 + c * 16 + k];
            const float m = M[c * 16 + k];
            f2 += q; cross += m * s; m2 += m * m;
        }
        const float ss  = f2 - 2.0f * cross + m2 * cnt;
        const float mse = ss / (safe * (float)C_CH);
        vsum = fmaxf(mse - DELTA_VAR, 0.0f);
        nsum = sqrtf(m2);
    }
    for (int idx = tid; idx < N_IMG * K_CL * K_CL; idx += 256) {
        const int n = idx >> 8, a = (idx >> 4) & 15, b = idx & 15;
        if (a != b) {
            const float* M = out + 4 + n * 512;
            float sq = 0.0f;
            for (int c = 0; c < C_CH; ++c) {
                const float d = M[c * 16 + a] - M[c * 16 + b];
                sq += d * d;
            }
            dsum += fmaxf(2.0f * DELTA_DIST - sqrtf(sq), 0.0f);
        }
    }
    redv[tid] = vsum; redd[tid] = dsum; redn[tid] = nsum;
    __syncthreads();
    if (tid == 0) {
        float v = 0.0f, d = 0.0f, nn = 0.0f;
        for (int i = 0; i < 256; ++i) { v += redv[i]; d += redd[i]; nn += redn[i]; }
        const float inv = 1.0f / (float)(N_IMG * K_CL);
        v *= inv; d *= inv; nn *= inv;
        out[1] = v; out[2] = d; out[3] = nn;
        out[0] = v + d + GAMMA_W * nn;               // alpha=beta=1, gamma=1e-3
    }
}

// ---------------------------------------------------------------------------
extern "C" void kernel_launch(void* const* d_in, const int* in_sizes, int n_in,
                              void* d_out, int out_size, void* d_ws, size_t ws_size,
                              hipStream_t stream) {
    (void)in_sizes; (void)n_in; (void)out_size; (void)ws_size;
    const float* feat = (const float*)d_in[0];
    const int*   gt   = (const int*)d_in[1];
    float* out      = (float*)d_out;
    float* partials = (float*)d_ws;                                  // 512*1040 f32 (~2.1MB)
    float* finals   = partials + (long)TOTAL_WGS * PART_STRIDE;      // 8*1040 f32

    cluster_phase1<<<dim3(TOTAL_WGS), dim3(256), 0, stream>>>(feat, gt, partials);
    cluster_phase2<<<dim3(N_IMG),     dim3(256), 0, stream>>>(partials, finals, out);
    cluster_phase3<<<dim3(1),         dim3(256), 0, stream>>>(finals, out);
}